// StandardGCN_85985245266465
// MI455X (gfx1250) — compile-verified
//
#include <hip/hip_runtime.h>
#include <hip/hip_bf16.h>
#include <math.h>

typedef float v2f __attribute__((ext_vector_type(2)));
typedef float v8f __attribute__((ext_vector_type(8)));

#define N_NODES 100000
#define DPAD    100096   // dinv region padded to 256B multiple (floats)

// ---------------------------------------------------------------- utilities
__global__ void fill_f32_kernel(float* __restrict__ p, float v, long long n) {
    long long t = (long long)blockIdx.x * blockDim.x + threadIdx.x;
    if (t < n) p[t] = v;
}

__global__ void deg_accum_kernel(const int* __restrict__ col, float* __restrict__ deg, int E) {
    int e = blockIdx.x * blockDim.x + threadIdx.x;
    if (e < E) atomicAdd(&deg[col[e]], 1.0f);
}

__global__ void rsqrt_ip_kernel(float* __restrict__ p, int n) {
    int i = blockIdx.x * blockDim.x + threadIdx.x;
    if (i < n) p[i] = rsqrtf(p[i]);   // deg >= 1 always (self loops)
}

// ---------------------------------------------------------------- WMMA GEMM
// out[M x NCOLS] = A[M x 64] @ W[64 x NCOLS], fp32 via V_WMMA_F32_16X16X4_F32.
// One wave computes one 16x16 output tile; K=64 -> 16 chained WMMAs.
// NCOLS is a template parameter so all strides are immediates.
// For the ragged last column tile (NCOLS=40), B addresses are CLAMPED instead
// of masked: garbage in lanes n>=NCOLS only affects D entries in those same
// lanes (column n of D depends only on column n of B), which are never stored.
template <int NCOLS>
__global__ void gemm_wmma_f32_kernel(const float* __restrict__ A,
                                     const float* __restrict__ W,
                                     float* __restrict__ out) {
    constexpr int K = 64;
    const int lane = threadIdx.x;          // 0..31
    const int m    = lane & 15;            // row within tile (A) / col within tile (B,D)
    const int hi   = lane >> 4;            // 0: K pair {0,1}; 1: K pair {2,3}
    const int row0 = blockIdx.x * 16;
    const int col0 = blockIdx.y * 16;
    const int n    = col0 + m;
    const int nsafe = (n < NCOLS) ? n : (NCOLS - 1);   // clamp, never branch

    v8f acc = {0.f, 0.f, 0.f, 0.f, 0.f, 0.f, 0.f, 0.f};

    const float* arow = A + (size_t)(row0 + m) * K;
    const float* wcol = W + nsafe;
    #pragma unroll
    for (int kb = 0; kb < K; kb += 4) {
        const int k = kb + 2 * hi;
        v2f a, b;
        // A 16x4 tile: lanes 0-15 -> K={kb,kb+1}, lanes 16-31 -> K={kb+2,kb+3}
        a.x = arow[k];
        a.y = arow[k + 1];
        // B 4x16 tile: VGPR r, lanes 0-15 -> K=kb+r, lanes 16-31 -> K=kb+r+2
        b.x = wcol[(size_t)k * NCOLS];
        b.y = wcol[(size_t)(k + 1) * NCOLS];
        acc = __builtin_amdgcn_wmma_f32_16x16x4_f32(
                /*neg_a=*/false, a, /*neg_b=*/false, b,
                /*c_mod=*/(short)0, acc, /*reuse_a=*/false, /*reuse_b=*/false);
    }

    if (n < NCOLS) {
        #pragma unroll
        for (int r = 0; r < 8; ++r) {
            const int mm = r + 8 * hi;     // D layout: VGPR r, lanes>=16 -> M=r+8
            out[(size_t)(row0 + mm) * NCOLS + n] = acc[r];
        }
    }
}

// ---------------------------------------------------------------- aggregation
// One wave per edge: agg[col[e]][f] += h[row[e]][f] * dinv[row[e]] * dinv[col[e]].
// Index/dinv loads are wave-uniform (one request); feature loads/atomics are
// coalesced; the 25.6MB destination lives in the 192MB L2 so atomics hit L2.
template <int D>
__global__ void edge_agg_kernel(const float* __restrict__ h,
                                const float* __restrict__ dinv,
                                const int* __restrict__ row,
                                const int* __restrict__ col,
                                float* __restrict__ agg, int E) {
    const int wavesPerBlock = blockDim.x >> 5;
    const int e = blockIdx.x * wavesPerBlock + (threadIdx.x >> 5);
    if (e >= E) return;
    const int lane = threadIdx.x & 31;
    const int r = row[e];
    const int c = col[e];
    const float nrm = dinv[r] * dinv[c];
    const float* __restrict__ hr = h + (size_t)r * D;
    float* __restrict__ ac = agg + (size_t)c * D;
    #pragma unroll
    for (int f = lane; f < D; f += 32)
        atomicAdd(&ac[f], hr[f] * nrm);
}

// self-loop contribution + bias + ReLU, h written in place of agg
__global__ void node_relu_kernel(float* __restrict__ agg,          // in: agg, out: h
                                 const float* __restrict__ hlin,   // pre-aggregation x@W
                                 const float* __restrict__ dinv,
                                 const float* __restrict__ bias,
                                 int N) {
    const int D = 64;
    long long t = (long long)blockIdx.x * blockDim.x + threadIdx.x;
    if (t >= (long long)N * D) return;
    int i = (int)(t >> 6);
    int f = (int)(t & 63);
    float d2 = dinv[i] * dinv[i];
    float v = agg[t] + hlin[t] * d2 + bias[f];
    agg[t] = fmaxf(v, 0.0f);
}

// self-loop + bias + log_softmax over D=40 classes, one thread per node
__global__ void final_lsm_kernel(const float* __restrict__ agg,
                                 const float* __restrict__ zlin,
                                 const float* __restrict__ dinv,
                                 const float* __restrict__ bias,
                                 float* __restrict__ out, int N) {
    int i = blockIdx.x * blockDim.x + threadIdx.x;
    if (i >= N) return;
    const int D = 40;
    float d2 = dinv[i] * dinv[i];
    float v[D];
    float mx = -INFINITY;
    #pragma unroll
    for (int f = 0; f < D; ++f) {
        float t = agg[(size_t)i * D + f] + zlin[(size_t)i * D + f] * d2 + bias[f];
        v[f] = t;
        mx = fmaxf(mx, t);
    }
    float s = 0.f;
    #pragma unroll
    for (int f = 0; f < D; ++f) s += expf(v[f] - mx);
    float lse = mx + logf(s);
    #pragma unroll
    for (int f = 0; f < D; ++f) out[(size_t)i * D + f] = v[f] - lse;
}

// ---------------------------------------------------------------- launch
extern "C" void kernel_launch(void* const* d_in, const int* in_sizes, int n_in,
                              void* d_out, int out_size, void* d_ws, size_t ws_size,
                              hipStream_t stream) {
    const float* x  = (const float*)d_in[0];   // [N, 64]
    const int*   ei = (const int*)d_in[1];     // [2, E]
    const float* W1 = (const float*)d_in[2];   // [64, 64]
    const float* b1 = (const float*)d_in[3];   // [64]
    const float* W2 = (const float*)d_in[4];   // [64, 40]
    const float* b2 = (const float*)d_in[5];   // [40]
    float* out = (float*)d_out;                // [N, 40]

    const int N = N_NODES;
    const int E = in_sizes[1] / 2;             // 1,600,000
    const int* row = ei;                       // edge_index[0] = sources (gather)
    const int* col = ei + E;                   // edge_index[1] = targets (scatter)

    float* ws   = (float*)d_ws;
    float* dinv = ws;                              // N floats (deg -> dinv in place)
    float* bufA = ws + DPAD;                       // N*64 floats: h1, then z (N*40)
    float* bufB = bufA + (size_t)N * 64;           // N*64 floats: agg1/h, then agg2

    const int TB = 256;
    const int wavesPerBlock = TB / 32;
    const int edgeBlocks = (E + wavesPerBlock - 1) / wavesPerBlock;

    // degree = 1 (self loop) + atomic count over targets; then rsqrt in place
    fill_f32_kernel<<<(N + TB - 1) / TB, TB, 0, stream>>>(dinv, 1.0f, N);
    deg_accum_kernel<<<(E + TB - 1) / TB, TB, 0, stream>>>(col, dinv, E);
    rsqrt_ip_kernel<<<(N + TB - 1) / TB, TB, 0, stream>>>(dinv, N);

    // ---- layer 1: h1 = x @ W1  (WMMA fp32)
    {
        dim3 g(N / 16, 64 / 16);
        gemm_wmma_f32_kernel<64><<<g, 32, 0, stream>>>(x, W1, bufA);
    }
    long long tot64 = (long long)N * 64;
    fill_f32_kernel<<<(int)((tot64 + TB - 1) / TB), TB, 0, stream>>>(bufB, 0.0f, tot64);
    edge_agg_kernel<64><<<edgeBlocks, TB, 0, stream>>>(bufA, dinv, row, col, bufB, E);
    node_relu_kernel<<<(int)((tot64 + TB - 1) / TB), TB, 0, stream>>>(
        bufB, bufA, dinv, b1, N);

    // ---- layer 2: z = h @ W2  (WMMA fp32, 40 cols -> 3 tiles, clamped B)
    {
        dim3 g(N / 16, 3);
        gemm_wmma_f32_kernel<40><<<g, 32, 0, stream>>>(bufB, W2, bufA);
    }
    long long tot40 = (long long)N * 40;
    fill_f32_kernel<<<(int)((tot40 + TB - 1) / TB), TB, 0, stream>>>(bufB, 0.0f, tot40);
    edge_agg_kernel<40><<<edgeBlocks, TB, 0, stream>>>(bufA, dinv, row, col, bufB, E);
    final_lsm_kernel<<<(N + TB - 1) / TB, TB, 0, stream>>>(bufB, bufA, dinv, b2, out, N);
}